// LSTMUnit_85452669321849
// MI455X (gfx1250) — compile-verified
//
#include <hip/hip_runtime.h>

typedef __attribute__((ext_vector_type(2))) float v2f;
typedef __attribute__((ext_vector_type(8))) float v8f;

#define X_DIM      2048
#define C_DIM      2048
#define I_DIM      4096              // X_DIM + C_DIM
#define ROWS_TOTAL (4 * C_DIM)       // 8192 fused gate rows: [f, i, c~, o]
#define KCHUNKS    8
#define KC         (I_DIM / KCHUNKS) // 512 K per chunk
#define WAVES_PER_WG  8
#define ROWS_PER_WAVE 16
#define ROWS_PER_WG   (WAVES_PER_WG * ROWS_PER_WAVE)  // 128
#define ROWBLOCKS     (ROWS_TOTAL / ROWS_PER_WG)      // 64

// ---------------------------------------------------------------------------
// GEMV via V_WMMA_F32_16X16X4_F32.
// Each wave: 16 rows of one gate matrix, one K-chunk of 512.
//   A (16x4 f32): lane L holds W[r0 + L%16][k + 2*(L/16) + {0,1}]  (b64 load)
//   B (4x16 f32): column-broadcast of i[k..k+3]:
//                 lane L holds i[k + 2*(L/16) + {0,1}]             (LDS read)
//   D: every column identical; acc[v] on lane L = partial for row v + 8*(L/16).
// ---------------------------------------------------------------------------
__global__ __launch_bounds__(256)
void lstm_gemv_wmma_k(const float* __restrict__ x, const float* __restrict__ h,
                      const float* __restrict__ wf, const float* __restrict__ wi,
                      const float* __restrict__ wc, const float* __restrict__ wo,
                      float* __restrict__ g_part)
{
    __shared__ float s_i[KC];

    const int tid  = threadIdx.x;
    const int wave = tid >> 5;
    const int lane = tid & 31;
    const int rb   = blockIdx.x;        // row block 0..63
    const int kc   = blockIdx.y;        // K chunk  0..7
    const int k0   = kc * KC;

    // Stage this K-chunk of i = concat(x, h) into LDS (chunk never straddles).
    const float* seg = (k0 < X_DIM) ? (x + k0) : (h + (k0 - X_DIM));
    for (int t = tid; t < KC; t += 256) s_i[t] = seg[t];
    __syncthreads();

    const int r0   = rb * ROWS_PER_WG + wave * ROWS_PER_WAVE;  // global gate row base
    const int gate = r0 >> 11;                                 // / 2048
    const int rg   = r0 & (C_DIM - 1);
    const float* W = (gate == 0) ? wf : (gate == 1) ? wi : (gate == 2) ? wc : wo;

    const int half = lane >> 4;     // K-phase half (A: lanes 16-31 hold K=2,3)
    const int m    = lane & 15;     // row within tile
    const float* wrow = W + (size_t)(rg + m) * I_DIM + k0 + 2 * half;
    const float* si   = s_i + 2 * half;

    v8f acc = {};
    #pragma unroll 2
    for (int k = 0; k < KC; k += 16) {
        // Four K=4 sub-steps; A cachelines for this row are fully consumed
        // inside this unrolled body (lane m covers K={0,1,4,5,8,9,12,13},
        // lane m+16 covers K={2,3,6,7,10,11,14,15}).
        v2f a0 = *(const v2f*)(wrow + k +  0);
        v2f a1 = *(const v2f*)(wrow + k +  4);
        v2f a2 = *(const v2f*)(wrow + k +  8);
        v2f a3 = *(const v2f*)(wrow + k + 12);
        v2f b0 = *(const v2f*)(si + k +  0);
        v2f b1 = *(const v2f*)(si + k +  4);
        v2f b2 = *(const v2f*)(si + k +  8);
        v2f b3 = *(const v2f*)(si + k + 12);
        acc = __builtin_amdgcn_wmma_f32_16x16x4_f32(false, a0, false, b0, (short)0, acc, false, false);
        acc = __builtin_amdgcn_wmma_f32_16x16x4_f32(false, a1, false, b1, (short)0, acc, false, false);
        acc = __builtin_amdgcn_wmma_f32_16x16x4_f32(false, a2, false, b2, (short)0, acc, false, false);
        acc = __builtin_amdgcn_wmma_f32_16x16x4_f32(false, a3, false, b3, (short)0, acc, false, false);
    }

    // Every accumulator column is identical (B was column-broadcast).
    // Lane 0 holds rows r0+0..7 in acc[0..7]; lane 16 holds rows r0+8..15.
    if (m == 0) {
        float* out = g_part + (size_t)kc * ROWS_TOTAL + r0 + 8 * half;
        float4 o0 = make_float4(acc[0], acc[1], acc[2], acc[3]);
        float4 o1 = make_float4(acc[4], acc[5], acc[6], acc[7]);
        *(float4*)(out + 0) = o0;
        *(float4*)(out + 4) = o1;
    }
}

// ---------------------------------------------------------------------------
// Fixed-order chunk reduction + bias + LSTM gate math (deterministic).
// d_out = [c_new (2048) | h_new (2048)]
// ---------------------------------------------------------------------------
__global__ __launch_bounds__(256)
void lstm_act(const float* __restrict__ g_part, const float* __restrict__ c_old,
              const float* __restrict__ bf, const float* __restrict__ bi,
              const float* __restrict__ bc, const float* __restrict__ bo,
              float* __restrict__ out)
{
    const int j = blockIdx.x * blockDim.x + threadIdx.x;
    if (j >= C_DIM) return;

    float gf = bf[j], gi = bi[j], gc = bc[j], go = bo[j];
    #pragma unroll
    for (int ch = 0; ch < KCHUNKS; ++ch) {
        const float* gp = g_part + (size_t)ch * ROWS_TOTAL;
        gf += gp[0 * C_DIM + j];
        gi += gp[1 * C_DIM + j];
        gc += gp[2 * C_DIM + j];
        go += gp[3 * C_DIM + j];
    }

    const float ff = 1.0f / (1.0f + expf(-gf));
    const float ii = 1.0f / (1.0f + expf(-gi));
    const float cc = tanhf(gc);
    const float oo = 1.0f / (1.0f + expf(-go));

    const float c_new = ff * c_old[j] + ii * cc;
    const float h_new = tanhf(c_new) * oo;

    out[j]         = c_new;   // c_new first (reference returns (c_new, h_new))
    out[C_DIM + j] = h_new;
}

extern "C" void kernel_launch(void* const* d_in, const int* in_sizes, int n_in,
                              void* d_out, int out_size, void* d_ws, size_t ws_size,
                              hipStream_t stream) {
    // setup_inputs() order: h, c, x, wf, bf, wi, bi, wo, bo, wc, bc
    const float* h  = (const float*)d_in[0];
    const float* c  = (const float*)d_in[1];
    const float* x  = (const float*)d_in[2];
    const float* wf = (const float*)d_in[3];
    const float* bf = (const float*)d_in[4];
    const float* wi = (const float*)d_in[5];
    const float* bi = (const float*)d_in[6];
    const float* wo = (const float*)d_in[7];
    const float* bo = (const float*)d_in[8];
    const float* wc = (const float*)d_in[9];
    const float* bc = (const float*)d_in[10];

    float* g_part = (float*)d_ws;   // KCHUNKS * 8192 floats = 256 KB, fully
                                    // rewritten every call (deterministic).

    dim3 grid(ROWBLOCKS, KCHUNKS);  // 64 x 8 workgroups, 8 waves each
    lstm_gemv_wmma_k<<<grid, 256, 0, stream>>>(x, h, wf, wi, wc, wo, g_part);
    lstm_act<<<(C_DIM + 255) / 256, 256, 0, stream>>>(g_part, c, bf, bi, bc, bo,
                                                      (float*)d_out);
    (void)in_sizes; (void)n_in; (void)out_size; (void)ws_size;
}